// BertCRF_14594298872013
// MI455X (gfx1250) — compile-verified
//
#include <hip/hip_runtime.h>

typedef __attribute__((ext_vector_type(2))) float v2f;
typedef __attribute__((ext_vector_type(4))) float v4f;
typedef __attribute__((ext_vector_type(8))) float v8f;

#define Bsz 64
#define Ssz 512
#define Hsz 768
#define Lsz 9
#define Npad 16
#define Mrows (Bsz * Ssz)   // 32768 rows -> 2048 16-row tiles

// ---------------------------------------------------------------------------
// Kernel 1: emissions = hidden @ cls_w + cls_b  via V_WMMA_F32_16X16X4_F32.
// One wave32 per 16-row tile; 8 waves per block. Weights staged in LDS,
// transposed and zero-padded N: 9 -> 16, so the K-loop is branch-free.
//
// K-slot permutation: WMMA slots {0,1} = lanes 0-15 vgpr{0,1}, slots {2,3} =
// lanes 16-31 vgpr{0,1}. We map lane-half h to global K = k0 + 4h + {0,1}
// (wmma A) and k0 + 4h + {2,3} (wmma B) identically for A and B fragments, so
// one b128 load per matrix feeds two WMMAs over 8 K values.
// ---------------------------------------------------------------------------
__global__ __launch_bounds__(256) void emis_gemm_wmma(
    const float* __restrict__ hidden, const float* __restrict__ w,
    const float* __restrict__ bias, float* __restrict__ em)
{
    __shared__ float sWT[Npad * Hsz];   // 48 KB: sWT[c*768 + k] = w[k][c], c>=9 -> 0

    const int tid  = threadIdx.x;
    const int lane = tid & 31;
    const int wave = tid >> 5;
    const int tile = blockIdx.x * 8 + wave;      // 2048 tiles total
    const int col  = lane & 15;                  // N (B/C), M (A)
    const int hb   = (lane >> 4) << 2;           // {0,4}: K base offset per lane-half

    // stage transposed, padded weights (uniform branch on c)
    for (int c = 0; c < Npad; ++c) {
        if (c < Lsz) {
            for (int k = tid; k < Hsz; k += 256) sWT[c * Hsz + k] = w[k * Lsz + c];
        } else {
            for (int k = tid; k < Hsz; k += 256) sWT[c * Hsz + k] = 0.f;
        }
    }
    __syncthreads();

    const float* arow = hidden + (size_t)(tile * 16 + col) * Hsz;
    const float* brow = &sWT[col * Hsz];

    v8f acc0 = {};
    v8f acc1 = {};

    #pragma unroll 4
    for (int k0 = 0; k0 < Hsz; k0 += 8) {
        v4f av = *(const v4f*)(arow + k0 + hb);   // global_load_b128
        v4f bv = *(const v4f*)(brow + k0 + hb);   // ds_load_b128 (16B aligned)

        v2f a0 = {av.x, av.y};
        v2f b0 = {bv.x, bv.y};
        acc0 = __builtin_amdgcn_wmma_f32_16x16x4_f32(
                   false, a0, false, b0, (short)0, acc0, false, false);

        v2f a1 = {av.z, av.w};
        v2f b1 = {bv.z, bv.w};
        acc1 = __builtin_amdgcn_wmma_f32_16x16x4_f32(
                   false, a1, false, b1, (short)0, acc1, false, false);
    }

    if (col < Lsz) {
        const float bv    = bias[col];
        const int   rbase = tile * 16 + ((lane >> 4) << 3);
        #pragma unroll
        for (int v = 0; v < 8; ++v) {
            em[(size_t)(rbase + v) * Lsz + col] = acc0[v] + acc1[v] + bv;
        }
    }
}

// ---------------------------------------------------------------------------
// Kernel 2: per-batch CRF negative log-likelihood terms.
// One wave32 per batch. Lane j owns forward-scan state j (j<9); numerator is
// strided-parallel over t then wave-reduced. Writes llh[b] = num - denom.
// ---------------------------------------------------------------------------
__global__ __launch_bounds__(32) void crf_forward(
    const float* __restrict__ em, const float* __restrict__ startT,
    const float* __restrict__ endT, const float* __restrict__ trans,
    const int* __restrict__ labels, const int* __restrict__ amask,
    float* __restrict__ llh)
{
    const int b    = blockIdx.x;
    const int lane = threadIdx.x;

    __shared__ float sT[Lsz * Lsz];
    for (int i = lane; i < Lsz * Lsz; i += 32) sT[i] = trans[i];
    __syncthreads();

    const float* emb = em + (size_t)b * Ssz * Lsz;
    const int*   tg  = labels + b * Ssz;
    const int*   mk  = amask + b * Ssz;

    // ---- numerator partials (t >= 1) and mask count, parallel over t ----
    float numPart = 0.f;
    int   mcnt    = 0;
    for (int t = lane; t < Ssz; t += 32) {
        const int m = mk[t];
        mcnt += (m != 0);
        if (t >= 1 && m) {
            int tp = tg[t - 1]; tp = (tp < 0 || tp >= Lsz) ? 0 : tp;
            int tc = tg[t];     tc = (tc < 0 || tc >= Lsz) ? 0 : tc;
            numPart += sT[tp * Lsz + tc] + emb[t * Lsz + tc];
        }
    }
    #pragma unroll
    for (int off = 16; off > 0; off >>= 1) {
        numPart += __shfl_down(numPart, off, 32);
        mcnt    += __shfl_down(mcnt, off, 32);
    }
    const float numTot = __shfl(numPart, 0, 32);
    const int   seqEnd = __shfl(mcnt, 0, 32) - 1;

    // ---- forward scan: lane j = state j; lanes 9..31 shadow state 8 ----
    const int jc = (lane < Lsz) ? lane : (Lsz - 1);
    float Tc[Lsz];
    #pragma unroll
    for (int i = 0; i < Lsz; ++i) Tc[i] = sT[i * Lsz + jc];

    float score = startT[jc] + emb[jc];
    for (int t = 1; t < Ssz; ++t) {
        float v[Lsz];
        float m = -INFINITY;
        #pragma unroll
        for (int i = 0; i < Lsz; ++i) {
            v[i] = __shfl(score, i, 32) + Tc[i];
            m = fmaxf(m, v[i]);
        }
        float s = 0.f;
        #pragma unroll
        for (int i = 0; i < Lsz; ++i) s += __expf(v[i] - m);
        const float nxt = m + __logf(s) + emb[t * Lsz + jc];
        if (mk[t]) score = nxt;
    }

    // ---- denominator: logsumexp over states of (score + end_transitions) ----
    float fin  = (lane < Lsz) ? (score + endT[jc]) : -INFINITY;
    float mall = fin;
    #pragma unroll
    for (int off = 16; off > 0; off >>= 1)
        mall = fmaxf(mall, __shfl_xor(mall, off, 32));
    float e = (lane < Lsz) ? __expf(fin - mall) : 0.f;
    #pragma unroll
    for (int off = 16; off > 0; off >>= 1)
        e += __shfl_xor(e, off, 32);
    const float denom = mall + __logf(e);

    if (lane == 0) {
        int t0 = tg[0];      t0 = (t0 < 0 || t0 >= Lsz) ? 0 : t0;
        int lt = tg[seqEnd]; lt = (lt < 0 || lt >= Lsz) ? 0 : lt;
        const float num = startT[t0] + emb[t0] + numTot + endT[lt];
        llh[b] = num - denom;
    }
}

// ---------------------------------------------------------------------------
// Kernel 3: loss = -mean(llh) over 64 batches -> d_out[0]
// ---------------------------------------------------------------------------
__global__ __launch_bounds__(32) void crf_loss(
    const float* __restrict__ llh, float* __restrict__ out)
{
    const int lane = threadIdx.x;
    float s = llh[lane] + llh[lane + 32];
    #pragma unroll
    for (int off = 16; off > 0; off >>= 1) s += __shfl_down(s, off, 32);
    if (lane == 0) out[0] = -s * (1.0f / (float)Bsz);
}

// ---------------------------------------------------------------------------
extern "C" void kernel_launch(void* const* d_in, const int* in_sizes, int n_in,
                              void* d_out, int out_size, void* d_ws, size_t ws_size,
                              hipStream_t stream)
{
    (void)in_sizes; (void)n_in; (void)out_size; (void)ws_size;

    const float* hidden = (const float*)d_in[0];   // (B,S,H)
    const float* cls_w  = (const float*)d_in[1];   // (H,L)
    const float* cls_b  = (const float*)d_in[2];   // (L,)
    const float* startT = (const float*)d_in[3];   // (L,)
    const float* endT   = (const float*)d_in[4];   // (L,)
    const float* trans  = (const float*)d_in[5];   // (L,L)
    const int*   labels = (const int*)d_in[6];     // (B,S)
    const int*   amask  = (const int*)d_in[7];     // (B,S)

    float* out = (float*)d_out;        // [0]=loss, [1..]=emissions (B*S*L)
    float* em  = out + 1;
    float* llh = (float*)d_ws;         // 64 floats scratch

    // 2048 tiles of 16 rows, 8 waves (256 threads) per block -> 256 blocks
    emis_gemm_wmma<<<Mrows / 16 / 8, 256, 0, stream>>>(hidden, cls_w, cls_b, em);
    crf_forward<<<Bsz, 32, 0, stream>>>(em, startT, endT, trans, labels, amask, llh);
    crf_loss<<<1, 32, 0, stream>>>(llh, out);
}